// EquisolidDistortion_86457691669204
// MI455X (gfx1250) — compile-verified
//
#include <hip/hip_runtime.h>

typedef float v4f __attribute__((ext_vector_type(4)));
typedef int   v4i __attribute__((ext_vector_type(4)));

// Geometry (from reference):
//   input  image : (3, 2048, 2048) f32
//   stage-1 grid : (1024, 4096) -> LUT: gather index or -1
//   output       : (3, 4096, 4096) f32
#define CH_IN   4194304     // 2048*2048
#define CH_OUT  16777216    // 4096*4096
#define MAP_N   4194304     // 1024*4096

// ---- minimax atan on [0,1], max err ~1.1e-5 rad (index quantum 2.4e-4) ------
__device__ __forceinline__ float atan_poly01(float t) {
    const float u = t * t;
    float p = fmaf(u, -0.01172120f, 0.05265332f);
    p = fmaf(u, p, -0.11643287f);
    p = fmaf(u, p,  0.19354346f);
    p = fmaf(u, p, -0.33262347f);
    p = fmaf(u, p,  0.99997726f);
    return p * t;
}

// atan(min/max) mapped to [0, pi/2]: one v_rcp_f32 + poly, no div chains.
__device__ __forceinline__ float atan_octant(float num, float den) {
    const float mx = fmaxf(num, den), mn = fminf(num, den);
    const float t  = mn * __builtin_amdgcn_rcpf(mx);
    const float a  = atan_poly01(t);
    return (num > den) ? (1.57079637f - a) : a;
}

// Stage-1 fisheye index at quantized grid point: py2*2048+px2, or -1 if masked.
__device__ __forceinline__ int stage1_index(int py1, int px1) {
    const float theta1 = (float)py1 * 1.5354803e-3f;      // (pi/2)/1023
    const float rad    = theta1 * 651.8986469f;           // * 2048/pi
    // v_sin/v_cos take revolutions: phi1/(2pi) = px1/4095 - 0.5
    const float u  = fmaf((float)px1, 2.4420024e-4f, -0.5f);
    const float cp = __builtin_amdgcn_cosf(u);
    const float sp = __builtin_amdgcn_sinf(u);
    const int px2 = (int)fmaf(rad, cp, 1024.0f);
    const int py2 = (int)fmaf(rad, sp, 1024.0f);
    const bool in = (rad <= 1024.0f) &
                    ((unsigned)px2 < 2048u) & ((unsigned)py2 < 2048u);
    return in ? (py2 * 2048 + px2) : -1;
}

// ---- pass 1: build 16MB stage-1 LUT (stays L2-resident for pass 2) ----------
__global__ __launch_bounds__(256)
void build_stage1_map_kernel(int* __restrict__ map)
{
    const int tid  = blockIdx.x * 256 + threadIdx.x;
    const int e0   = tid << 2;              // 4 entries, same theta row
    const int py1  = e0 >> 12;
    const int px1b = e0 & 4095;
    v4i m;
#pragma unroll
    for (int k = 0; k < 4; ++k) m[k] = stage1_index(py1, px1b + k);
    *(v4i*)(map + e0) = m;                  // regular store: keep hot in L2
}

// ---- pass 2: 4-fold mirror symmetry; one atan pair serves 4 output pixels ---
template <bool USE_LUT>
__global__ __launch_bounds__(256)
void equisolid_sym_kernel(const float* __restrict__ img,
                          const int*   __restrict__ map,
                          float*       __restrict__ out)
{
    const int tid = blockIdx.x * 256 + threadIdx.x;
    const int i   = tid >> 9;               // top row      0..2047
    const int j0  = (tid & 511) << 2;       // left col base 0..2044
    const int iB  = 4095 - i;               // mirrored row
    const int jR0 = 4092 - j0;              // mirrored col base (of 4-group)

    const float yT = fmaf((float)i, 2.0f / 4095.0f, -1.0f);   // strictly < 0
    const float ay = fabsf(yT);
    const float yy = yT * yT;

    // plain register arrays (fully unrolled indices); packed to v4f at store
    float tlr[4], tlg[4], tlb[4], trr[4], trg[4], trb[4];
    float blr[4], blg[4], blb[4], brr[4], brg[4], brb[4];

#pragma unroll
    for (int k = 0; k < 4; ++k) {
        const float xT = fmaf((float)(j0 + k), 2.0f / 4095.0f, -1.0f); // < 0
        const float ax = fabsf(xT);
        const float r2 = fmaf(xT, xT, yy);
        const bool  valid = (r2 <= 1.0f);

        // theta2 = 2*asin(rn/sqrt2): cos=1-r2, sin=rn*sqrt(2-r2); shared by all 4
        const float rn = sqrtf(r2);
        const float sn = rn * sqrtf(fmaxf(2.0f - r2, 0.0f));
        const float cs = 1.0f - r2;
        const float th = atan_octant(sn, cs);           // [0, pi/2] in valid disk
        int py1 = (int)(th * 651.2620239f);             // *1023*(2/pi)
        py1 = min(max(py1, 0), 1023);

        // |phi| via ONE rcp+poly; four mirrors are +-aa, +-(pi-aa)
        const float aa = atan_octant(ay, ax);           // angle vs +x axis, [0,pi/2]
        const float phiTL = aa - 3.14159274f;           // x<0, y<0
        const float phiTR = -aa;                        // x>0, y<0
        const float phiBL = 3.14159274f - aa;           // x<0, y>0
        const float phiBR = aa;                         // x>0, y>0

        auto sample = [&](float phi, float& rr, float& gg, float& bb) {
            int px1 = (int)fmaf(phi, 651.7395020f, 2047.5f); // *4095/(2pi)+2047.5
            px1 = min(max(px1, 0), 4095);
            int s;
            if (USE_LUT) s = map[(py1 << 12) | px1];
            else         s = stage1_index(py1, px1);
            const bool m    = valid & (s >= 0);
            const int  sidx = max(s, 0);                 // always in-bounds gather
            const float c0 = img[sidx];
            const float c1 = img[CH_IN + sidx];
            const float c2 = img[2 * CH_IN + sidx];
            rr = m ? c0 : 0.0f;
            gg = m ? c1 : 0.0f;
            bb = m ? c2 : 0.0f;
        };

        sample(phiTL, tlr[k],     tlg[k],     tlb[k]);      // ascending cols
        sample(phiTR, trr[3 - k], trg[3 - k], trb[3 - k]);  // reversed cols
        sample(phiBL, blr[k],     blg[k],     blb[k]);
        sample(phiBR, brr[3 - k], brg[3 - k], brb[3 - k]);
    }

    // 201MB write-once output: NT b128 stores keep image(48MB)+LUT(16MB) in L2.
    const int oTL = i  * 4096 + j0,  oTR = i  * 4096 + jR0;
    const int oBL = iB * 4096 + j0,  oBR = iB * 4096 + jR0;
#define NT_STORE(a, off)                                                     \
    do {                                                                     \
        v4f _v = { (a)[0], (a)[1], (a)[2], (a)[3] };                         \
        __builtin_nontemporal_store(_v, (v4f*)(out + (off)));                \
    } while (0)
    NT_STORE(tlr, oTL);              NT_STORE(trr, oTR);
    NT_STORE(blr, oBL);              NT_STORE(brr, oBR);
    NT_STORE(tlg, CH_OUT + oTL);     NT_STORE(trg, CH_OUT + oTR);
    NT_STORE(blg, CH_OUT + oBL);     NT_STORE(brg, CH_OUT + oBR);
    NT_STORE(tlb, 2*CH_OUT + oTL);   NT_STORE(trb, 2*CH_OUT + oTR);
    NT_STORE(blb, 2*CH_OUT + oBL);   NT_STORE(brb, 2*CH_OUT + oBR);
#undef NT_STORE
}

extern "C" void kernel_launch(void* const* d_in, const int* in_sizes, int n_in,
                              void* d_out, int out_size, void* d_ws, size_t ws_size,
                              hipStream_t stream) {
    const float* img = (const float*)d_in[0];
    float*       out = (float*)d_out;

    const dim3 block(256);
    // 2048 rows x 512 col-groups = 1,048,576 threads; 16 px/thread
    const dim3 gridMain(2048 * 512 / 256);

    if (ws_size >= (size_t)MAP_N * sizeof(int)) {
        int* map = (int*)d_ws;
        build_stage1_map_kernel<<<dim3(MAP_N / 4 / 256), block, 0, stream>>>(map);
        equisolid_sym_kernel<true><<<gridMain, block, 0, stream>>>(img, map, out);
    } else {
        equisolid_sym_kernel<false><<<gridMain, block, 0, stream>>>(img, nullptr, out);
    }
}